// MLFF_dmirror_11390253269386
// MI455X (gfx1250) — compile-verified
//
#include <hip/hip_runtime.h>

// ---------------------------------------------------------------------------
// MLFF fitting-net fwd+bwd + force contraction for gfx1250 (MI455X).
//
// Shapes: B=8, N=512, M=100, F=42; MLP dims 42->128->64->32->1 (sigmoid, linear out)
// d_out = [ Etot(8) | Force(8*512*3) ]  (float32)
//
// Perf model: dfeat stream (206 MB, read once, non-temporal) dominates at
// 23.3 TB/s -> ~9 us. MLP is 0.2 GFLOP -> done with V_WMMA_F32_16X16X4_F32
// (full f32 precision on the matrix cores), fully unrolled K-chains.
// ---------------------------------------------------------------------------

typedef float v2f __attribute__((ext_vector_type(2)));
typedef float v8f __attribute__((ext_vector_type(8)));
typedef float f4  __attribute__((ext_vector_type(4)));

#define BB   8
#define NN   512
#define MM   100
#define FF   42
#define FP   44        // F padded to multiple of 4 for K-steps
#define H1   128
#define H2   64
#define H3   32
#define NATOM (BB*NN)  // 4096

// Fast sigmoid: v_exp_f32 + v_add + v_rcp_f32 (no IEEE div expansion).
__device__ __forceinline__ float sigmoidf(float x) {
  return __builtin_amdgcn_rcpf(1.0f + __expf(-x));
}

// One 16x16 f32 output tile: C = A[16,K] @ B[K,16] using V_WMMA_F32_16X16X4_F32.
// A in LDS, row-major, stride sA, zero-padded so K-steps of 4 are safe.
// ISA layouts (CDNA5 ISA): lane&15 = row (A) / col (B); kh = lane>>4 selects
// the K pair {2kh, 2kh+1} inside each K-step of 4. KSTEPS is compile-time so
// the whole fragment-load + WMMA chain unrolls and loads batch ahead of math.
template <int KSTEPS, typename GetB>
__device__ __forceinline__ v8f gemm_tile(const float* __restrict__ A, int sA,
                                         int kh, int row, int n, GetB getB) {
  v8f c = {};
#pragma unroll
  for (int ks = 0; ks < KSTEPS; ++ks) {
    const int k0 = ks * 4 + 2 * kh;
    v2f a, b;
    a.x = A[row * sA + k0];
    a.y = A[row * sA + k0 + 1];
    b.x = getB(k0, n);
    b.y = getB(k0 + 1, n);
    c = __builtin_amdgcn_wmma_f32_16x16x4_f32(false, a, false, b,
                                              (short)0, c, false, false);
  }
  return c;
}

// ---------------------------------------------------------------------------
// Kernel 1: per-atom MLP forward + backward. 1 wave = 16 atoms, 2 waves/block.
// Wave-private LDS regions -> no barriers needed; WMMA always full-EXEC.
// ---------------------------------------------------------------------------
__global__ __launch_bounds__(64) void mlp_fwd_bwd_kernel(
    const float* __restrict__ image,
    const float* __restrict__ W0, const float* __restrict__ b0,
    const float* __restrict__ W1, const float* __restrict__ b1,
    const float* __restrict__ W2, const float* __restrict__ b2,
    const float* __restrict__ W3, const float* __restrict__ b3,
    float* __restrict__ dEi,   // [NATOM, FF]
    float* __restrict__ Ei)    // [NATOM]
{
  __shared__ float xs [2][16 * FP];
  __shared__ float h1s[2][16 * H1];
  __shared__ float h2s[2][16 * H2];
  __shared__ float h3s[2][16 * H3];

  const int wave = threadIdx.x >> 5;
  const int lane = threadIdx.x & 31;
  const int kh   = lane >> 4;        // 0/1: which K/M half this lane owns
  const int row  = lane & 15;        // A-matrix row this lane owns
  const int col  = lane & 15;        // B/C column this lane owns
  const int atom0 = blockIdx.x * 32 + wave * 16;

  float* X  = xs [wave];
  float* h1 = h1s[wave];
  float* h2 = h2s[wave];
  float* h3 = h3s[wave];

  // ---- stage input features (zero-padded 42 -> 44) ----
  for (int t = lane; t < 16 * FP; t += 32) {
    const int r = t / FP, c = t - r * FP;
    X[t] = (c < FF) ? image[(atom0 + r) * FF + c] : 0.0f;
  }

  // ---- forward: h1 = sigmoid(X @ W0 + b0) ----
#pragma unroll
  for (int nt = 0; nt < H1 / 16; ++nt) {
    const int n = nt * 16 + col;
    v8f c = gemm_tile<FP / 4>(X, FP, kh, row, n,
                      [&](int k, int nn) { return (k < FF) ? W0[k * H1 + nn] : 0.0f; });
    const float bv = b0[n];
#pragma unroll
    for (int j = 0; j < 8; ++j) h1[(j + 8 * kh) * H1 + n] = sigmoidf(c[j] + bv);
  }
  // ---- h2 = sigmoid(h1 @ W1 + b1) ----
#pragma unroll
  for (int nt = 0; nt < H2 / 16; ++nt) {
    const int n = nt * 16 + col;
    v8f c = gemm_tile<H1 / 4>(h1, H1, kh, row, n,
                      [&](int k, int nn) { return W1[k * H2 + nn]; });
    const float bv = b1[n];
#pragma unroll
    for (int j = 0; j < 8; ++j) h2[(j + 8 * kh) * H2 + n] = sigmoidf(c[j] + bv);
  }
  // ---- h3 = sigmoid(h2 @ W2 + b2) ----
#pragma unroll
  for (int nt = 0; nt < H3 / 16; ++nt) {
    const int n = nt * 16 + col;
    v8f c = gemm_tile<H2 / 4>(h2, H2, kh, row, n,
                      [&](int k, int nn) { return W2[k * H3 + nn]; });
    const float bv = b2[n];
#pragma unroll
    for (int j = 0; j < 8; ++j) h3[(j + 8 * kh) * H3 + n] = sigmoidf(c[j] + bv);
  }

  // ---- Ei = h3 @ W3 + b3 (lanes 0..15, one atom each) ----
  if (lane < 16) {
    float e = b3[0];
#pragma unroll
    for (int j = 0; j < H3; ++j) e += h3[lane * H3 + j] * W3[j];
    Ei[atom0 + lane] = e;
  }

  // ---- backward: g3 = W3 .* h3 .* (1-h3)   (in place over h3) ----
#pragma unroll
  for (int c2 = 0; c2 < 16; ++c2) {
    const int cc = kh * 16 + c2;
    const float v = h3[row * H3 + cc];
    h3[row * H3 + cc] = W3[cc] * v * (1.0f - v);
  }
  // ---- g2 = (g3 @ W2^T) .* h2 .* (1-h2)    (in place over h2) ----
#pragma unroll
  for (int nt = 0; nt < H2 / 16; ++nt) {
    const int n = nt * 16 + col;
    v8f c = gemm_tile<H3 / 4>(h3, H3, kh, row, n,
                      [&](int k, int nn) { return W2[nn * H3 + k]; });
#pragma unroll
    for (int j = 0; j < 8; ++j) {
      const int r2 = j + 8 * kh;
      const float v = h2[r2 * H2 + n];
      h2[r2 * H2 + n] = c[j] * v * (1.0f - v);
    }
  }
  // ---- g1 = (g2 @ W1^T) .* h1 .* (1-h1)    (in place over h1) ----
#pragma unroll
  for (int nt = 0; nt < H1 / 16; ++nt) {
    const int n = nt * 16 + col;
    v8f c = gemm_tile<H2 / 4>(h2, H2, kh, row, n,
                      [&](int k, int nn) { return W1[nn * H2 + k]; });
#pragma unroll
    for (int j = 0; j < 8; ++j) {
      const int r2 = j + 8 * kh;
      const float v = h1[r2 * H1 + n];
      h1[r2 * H1 + n] = c[j] * v * (1.0f - v);
    }
  }
  // ---- dEi = g1 @ W0^T  (N=42 -> 3 tiles, guarded stores) ----
#pragma unroll
  for (int nt = 0; nt < 3; ++nt) {
    const int n = nt * 16 + col;
    v8f c = gemm_tile<H1 / 4>(h1, H1, kh, row, n,
                      [&](int k, int nn) { return (nn < FF) ? W0[nn * H1 + k] : 0.0f; });
    if (n < FF) {
#pragma unroll
      for (int j = 0; j < 8; ++j)
        dEi[(atom0 + j + 8 * kh) * FF + n] = c[j];
    }
  }
}

// ---------------------------------------------------------------------------
// Kernel 2: Etot[b] = sum_n Ei[b,n]  (deterministic tree reduction)
// ---------------------------------------------------------------------------
__global__ __launch_bounds__(256) void etot_kernel(const float* __restrict__ Ei,
                                                   float* __restrict__ Etot) {
  __shared__ float s[256];
  const int b = blockIdx.x, t = threadIdx.x;
  s[t] = Ei[b * NN + t] + Ei[b * NN + 256 + t];
  __syncthreads();
  for (int st = 128; st > 0; st >>= 1) {
    if (t < st) s[t] += s[t + st];
    __syncthreads();
  }
  if (t == 0) Etot[b] = s[0];
}

// ---------------------------------------------------------------------------
// Kernel 3: Force[b,n,:] = sum_m sum_f mask*dEi[b,idx(m),f]*dfeat[b,n,m,f,:]
// One block per (b,n): gather 4200-float g-tile to LDS (dEi is L2-hot, 688 KB),
// then stream this atom's 50.4 KB dfeat chunk with 128-bit non-temporal loads.
// ---------------------------------------------------------------------------
__global__ __launch_bounds__(256) void force_kernel(
    const float* __restrict__ dfeat,     // [NATOM, MM, FF, 3]
    const int*   __restrict__ neighbor,  // [NATOM, MM]
    const float* __restrict__ dEi,       // [NATOM, FF]
    float* __restrict__ outF)            // [NATOM, 3]
{
  __shared__ float g[MM * FF];           // 4200 floats
  __shared__ int   nb[MM];
  __shared__ float red0[256], red1[256], red2[256];

  const int atom = blockIdx.x;
  const int b    = atom >> 9;            // atom / 512
  const int t    = threadIdx.x;

  if (t < MM) nb[t] = neighbor[atom * MM + t];
  __syncthreads();

  for (int j = t; j < MM * FF; j += 256) {
    const int m = j / FF, f = j - m * FF;
    const int v = nb[m];
    g[j] = (v > 0) ? dEi[(b * NN + (v - 1)) * FF + f] : 0.0f;
  }
  __syncthreads();

  const float* df = dfeat + (size_t)atom * (MM * FF * 3);
  float a0 = 0.0f, a1 = 0.0f, a2 = 0.0f;

  for (int e = t * 4; e < MM * FF * 3; e += 256 * 4) {
    const f4 v = __builtin_nontemporal_load((const f4*)(df + e));
#pragma unroll
    for (int j = 0; j < 4; ++j) {
      const int ee = e + j;
      const int m  = ee / (FF * 3);
      const int r  = ee - m * (FF * 3);
      const int f  = r / 3;
      const int d  = r - 3 * f;
      const float val = g[m * FF + f] * v[j];
      if (d == 0)      a0 += val;
      else if (d == 1) a1 += val;
      else             a2 += val;
    }
  }

  red0[t] = a0; red1[t] = a1; red2[t] = a2;
  __syncthreads();
  for (int st = 128; st > 0; st >>= 1) {
    if (t < st) { red0[t] += red0[t + st]; red1[t] += red1[t + st]; red2[t] += red2[t + st]; }
    __syncthreads();
  }
  if (t == 0) {
    outF[atom * 3 + 0] = red0[0];
    outF[atom * 3 + 1] = red1[0];
    outF[atom * 3 + 2] = red2[0];
  }
}

// ---------------------------------------------------------------------------
extern "C" void kernel_launch(void* const* d_in, const int* in_sizes, int n_in,
                              void* d_out, int out_size, void* d_ws, size_t ws_size,
                              hipStream_t stream) {
  const float* image    = (const float*)d_in[0];
  const float* dfeat    = (const float*)d_in[1];
  const int*   neighbor = (const int*)  d_in[2];
  const float* W0 = (const float*)d_in[3];  const float* b0 = (const float*)d_in[4];
  const float* W1 = (const float*)d_in[5];  const float* b1 = (const float*)d_in[6];
  const float* W2 = (const float*)d_in[7];  const float* b2 = (const float*)d_in[8];
  const float* W3 = (const float*)d_in[9];  const float* b3 = (const float*)d_in[10];

  float* dEi = (float*)d_ws;                 // NATOM*FF floats
  float* Ei  = dEi + (size_t)NATOM * FF;     // NATOM floats
  float* out = (float*)d_out;                // [Etot(8) | Force(NATOM*3)]

  mlp_fwd_bwd_kernel<<<NATOM / 32, 64, 0, stream>>>(image, W0, b0, W1, b1,
                                                    W2, b2, W3, b3, dEi, Ei);
  etot_kernel<<<BB, 256, 0, stream>>>(Ei, out);
  force_kernel<<<NATOM, 256, 0, stream>>>(dfeat, neighbor, dEi, out + BB);
}